// HyperEConv_58282706207094
// MI455X (gfx1250) — compile-verified
//
#include <hip/hip_runtime.h>

// ---------------------------------------------------------------------------
// HyperEConv on MI455X (gfx1250, wave32)
//   y = X @ W^T + b  via V_WMMA_F32_16X16X4_F32 (exact fp32 matrix path)
//   segment_sum via L2-resident global_atomic_add_f32 (inline asm, guaranteed)
// ---------------------------------------------------------------------------

#define D 128
#define LPAD 132   // LDS row stride in floats: 132 % 64 = 4 -> conflict-free b64
                   // fragment reads across 16 lanes; 132*4 = 528 B (16B aligned)

typedef float v2f __attribute__((ext_vector_type(2)));
typedef float v8f __attribute__((ext_vector_type(8)));

// ---------------------------------------------------------------------------
// GEMM: Y[M,128] = X[M,128] @ W[128,128]^T + bias[128]
// Block = 256 threads (8 waves). Each block: 64 rows of output.
//   wave n (0..7) owns output columns [16n, 16n+16), 4 M-tiles of 16 rows.
// WMMA fragment layouts (ISA 7.12.2, 32-bit 16x16x4):
//   A 16x4 : lane l -> row l%16, k = 4*kb + 2*(l/16) + {0,1}   (float2)
//   B 4x16 : B[k][n] = W[n0+n][k] -> same indexing on W rows    (float2)
//   C/D    : vgpr i -> (M = i + 8*(l/16), N = l%16)
// ---------------------------------------------------------------------------
__global__ __launch_bounds__(256) void gemm128_bias_kernel(
    const float* __restrict__ X, const float* __restrict__ W,
    const float* __restrict__ bias, float* __restrict__ Y, int M)
{
    __shared__ float sW[128 * LPAD];  // full weight matrix, 67.6 KB
    __shared__ float sX[64 * LPAD];   // 64-row activation tile, 33.8 KB

    const int tid   = threadIdx.x;
    const int mbase = blockIdx.x * 64;

    // Stage W (128x128): 4096 float4, 16 per thread
    #pragma unroll
    for (int i = 0; i < 16; ++i) {
        const int q = i * 256 + tid;     // float4 index
        const int r = q >> 5;            // 32 float4 per row
        const int c = q & 31;
        const float4 v = *reinterpret_cast<const float4*>(W + r * D + c * 4);
        *reinterpret_cast<float4*>(&sW[r * LPAD + c * 4]) = v;
    }
    // Stage X tile (64 rows, clamped at M-1): 2048 float4, 8 per thread
    #pragma unroll
    for (int i = 0; i < 8; ++i) {
        const int q = i * 256 + tid;
        const int r = q >> 5;
        const int c = q & 31;
        int row = mbase + r;
        row = row < M ? row : (M - 1);
        const float4 v = *reinterpret_cast<const float4*>(X + (size_t)row * D + c * 4);
        *reinterpret_cast<float4*>(&sX[r * LPAD + c * 4]) = v;
    }
    __syncthreads();

    const int wv   = tid >> 5;       // wave id = N-tile
    const int lane = tid & 31;
    const int lq   = lane & 15;
    const int half = lane >> 4;
    const int fo   = 2 * half;       // fragment column offset within k-block

    const float b = bias[wv * 16 + lq];  // N depends only on lane
    v8f acc[4];
    #pragma unroll
    for (int mt = 0; mt < 4; ++mt) {
        v8f t = {b, b, b, b, b, b, b, b};
        acc[mt] = t;
    }

    const float* wrow = &sW[(wv * 16 + lq) * LPAD + fo];

    #pragma unroll 4
    for (int kb = 0; kb < 32; ++kb) {
        const v2f bf = *reinterpret_cast<const v2f*>(wrow + kb * 4);
        #pragma unroll
        for (int mt = 0; mt < 4; ++mt) {
            const v2f af = *reinterpret_cast<const v2f*>(
                &sX[(mt * 16 + lq) * LPAD + kb * 4 + fo]);
            acc[mt] = __builtin_amdgcn_wmma_f32_16x16x4_f32(
                /*neg_a=*/false, af, /*neg_b=*/false, bf,
                /*c_mod=*/(short)0, acc[mt],
                /*reuse_a=*/false, /*reuse_b=*/false);
        }
    }

    // Store: lanes 0-15 cover rows rbase..rbase+7, lanes 16-31 rows rbase+8..+15.
    // Fast path (all full tiles except at most the last block): unguarded
    // stores collapse to 32 global_store_b32 with immediate offsets.
    const int col = wv * 16 + lq;
    if (mbase + 64 <= M) {
        float* yb = Y + (size_t)(mbase + half * 8) * D + col;
        #pragma unroll
        for (int mt = 0; mt < 4; ++mt) {
            #pragma unroll
            for (int i = 0; i < 8; ++i) {
                yb[(size_t)(mt * 16 + i) * D] = acc[mt][i];
            }
        }
    } else {
        #pragma unroll
        for (int mt = 0; mt < 4; ++mt) {
            const int rbase = mbase + mt * 16 + half * 8;
            #pragma unroll
            for (int i = 0; i < 8; ++i) {
                const int row = rbase + i;
                if (row < M) Y[(size_t)row * D + col] = acc[mt][i];
            }
        }
    }
}

// ---------------------------------------------------------------------------
// Scatter: aggr[didx[e], :] += src[sidx[e], :]   (32 lanes x float4 per edge)
// Destination arrays (51 MB / 13 MB) are L2-resident (192 MB) -> hardware
// global_atomic_add_f32 resolves at L2 atomic throughput. Inline asm
// guarantees the single-instruction f32 atomic (no CAS-loop fallback);
// S_ENDPGM's implicit wait-idle covers the untracked STOREcnt.
// ---------------------------------------------------------------------------
__global__ __launch_bounds__(256) void scatter_add_kernel(
    const float* __restrict__ src, const int* __restrict__ sidx,
    const int* __restrict__ didx, float* __restrict__ aggr, int nE)
{
    const int  t = threadIdx.x;
    const long e = (long)blockIdx.x * 8 + (t >> 5);
    if (e >= nE) return;
    const int lane = t & 31;
    const int s = sidx[e];
    const int d = didx[e];
    const float4 v = *reinterpret_cast<const float4*>(src + (size_t)s * D + lane * 4);
    float* p = aggr + (size_t)d * D + lane * 4;
    asm volatile(
        "global_atomic_add_f32 %0, %1, off\n\t"
        "global_atomic_add_f32 %0, %2, off offset:4\n\t"
        "global_atomic_add_f32 %0, %3, off offset:8\n\t"
        "global_atomic_add_f32 %0, %4, off offset:12"
        :
        : "v"(p), "v"(v.x), "v"(v.y), "v"(v.z), "v"(v.w)
        : "memory");
}

// ---------------------------------------------------------------------------
// out = lin + aggr*lin = lin * (1 + aggr)      (float4 streaming)
// ---------------------------------------------------------------------------
__global__ __launch_bounds__(256) void update_mul_kernel(
    const float* __restrict__ lin, const float* __restrict__ aggr,
    float* __restrict__ out, long n4)
{
    const long i = (long)blockIdx.x * blockDim.x + threadIdx.x;
    if (i >= n4) return;
    const float4 l = reinterpret_cast<const float4*>(lin)[i];
    const float4 g = reinterpret_cast<const float4*>(aggr)[i];
    float4 o;
    o.x = l.x * (1.0f + g.x);
    o.y = l.y * (1.0f + g.y);
    o.z = l.z * (1.0f + g.z);
    o.w = l.w * (1.0f + g.w);
    reinterpret_cast<float4*>(out)[i] = o;
}

// ---------------------------------------------------------------------------
extern "C" void kernel_launch(void* const* d_in, const int* in_sizes, int n_in,
                              void* d_out, int out_size, void* d_ws, size_t ws_size,
                              hipStream_t stream)
{
    const float* x   = (const float*)d_in[0];
    const float* w   = (const float*)d_in[1];
    const float* Wx1 = (const float*)d_in[2];
    const float* bx1 = (const float*)d_in[3];
    const float* Ww1 = (const float*)d_in[4];
    const float* bw1 = (const float*)d_in[5];
    const float* Wx2 = (const float*)d_in[6];
    const float* bx2 = (const float*)d_in[7];
    const float* Ww2 = (const float*)d_in[8];
    const float* bw2 = (const float*)d_in[9];
    const int*   h   = (const int*)d_in[10];

    const int nX = in_sizes[0] / D;
    const int nW = in_sizes[1] / D;
    const int nE = in_sizes[10] / 2;
    const int* vidx = h;        // h[0]
    const int* eidx = h + nE;   // h[1]

    // Workspace: xv/xe (nX*D) | tmpw (nW*D) | aggr (nX*D, reused for aggr_w)
    float* ws   = (float*)d_ws;
    float* xv   = ws;
    float* tmpw = xv + (size_t)nX * D;
    float* aggr = tmpw + (size_t)nW * D;

    // d_out = [w_new (nW*D) | x_new (nX*D)]  (reference returns (w_new, x_new))
    float* w_out = (float*)d_out;
    float* x_out = w_out + (size_t)nW * D;

    const dim3 blk(256);
    const int gX = (nX + 63) / 64;
    const int gW = (nW + 63) / 64;
    const int gE = (nE + 7) / 8;
    const long n4X = (long)nX * D / 4;
    const long n4W = (long)nW * D / 4;

    // ---- MPToVertex ----
    hipMemsetAsync(aggr, 0, (size_t)nX * D * sizeof(float), stream);
    gemm128_bias_kernel<<<gX, blk, 0, stream>>>(x, Wx1, bx1, xv, nX);
    gemm128_bias_kernel<<<gW, blk, 0, stream>>>(w, Ww1, bw1, tmpw, nW);
    scatter_add_kernel<<<gE, blk, 0, stream>>>(tmpw, eidx, vidx, aggr, nE);
    update_mul_kernel<<<(int)((n4X + 255) / 256), blk, 0, stream>>>(xv, aggr, x_out, n4X);

    // ---- MPToEdge ----
    hipMemsetAsync(aggr, 0, (size_t)nW * D * sizeof(float), stream);
    gemm128_bias_kernel<<<gX, blk, 0, stream>>>(x_out, Wx2, bx2, xv, nX);   // xe
    gemm128_bias_kernel<<<gW, blk, 0, stream>>>(w, Ww2, bw2, tmpw, nW);     // we
    scatter_add_kernel<<<gE, blk, 0, stream>>>(xv, vidx, eidx, aggr, nE);
    update_mul_kernel<<<(int)((n4W + 255) / 256), blk, 0, stream>>>(tmpw, aggr, w_out, n4W);
}